// OutputBlock_Dropout_944892805680
// MI455X (gfx1250) — compile-verified
//
#include <hip/hip_runtime.h>
#include <math.h>

typedef float v2f __attribute__((ext_vector_type(2)));
typedef float v8f __attribute__((ext_vector_type(8)));

#define N_PARTICLES 50000
#define N_EDGES     800000
#define EMBED       128
#define NRBF        16
#define OUT_EMBED   256

#define ROWS_PER_WG 64
#define S_STRIDE    129   // EMBED + 1   (odd -> conflict-free LDS A-frag reads)
#define H_STRIDE    257   // OUT_EMBED+1 (odd -> conflict-free)

// ---------------------------------------------------------------------------
// 1) zero the segment-sum accumulator in workspace
// ---------------------------------------------------------------------------
__global__ __launch_bounds__(256) void zero_kernel(float4* p, int n4) {
    int i = blockIdx.x * 256 + threadIdx.x;
    if (i < n4) p[i] = make_float4(0.f, 0.f, 0.f, 0.f);
}

// ---------------------------------------------------------------------------
// 2) edge kernel: scaled = messages * (rbf @ W_rbf); scatter-add to summed
//    memory-bound (~464 MB reads); native f32 atomics into L2-resident 25.6MB
// ---------------------------------------------------------------------------
__global__ __launch_bounds__(256) void edge_scatter_kernel(
    const float* __restrict__ messages, const float* __restrict__ rbf,
    const int*   __restrict__ idx_i,    const float* __restrict__ W_rbf,
    float*       __restrict__ summed)
{
    __shared__ float wrbf[NRBF * EMBED];   // 8 KB
    const int tid = threadIdx.x;
    for (int p = tid; p < NRBF * EMBED; p += 256) wrbf[p] = W_rbf[p];
    __syncthreads();

    const int  j   = tid & (EMBED - 1);        // feature 0..127
    const int  sub = tid >> 7;                 // 0..1 (2 edges per block)
    const long e   = (long)blockIdx.x * 2 + sub;
    if (e >= N_EDGES) return;

    float t = 0.f;
#pragma unroll
    for (int k = 0; k < NRBF; ++k)
        t = fmaf(rbf[e * NRBF + k], wrbf[k * EMBED + j], t);

    const float val = messages[e * EMBED + j] * t;
    __hip_atomic_fetch_add(&summed[(long)idx_i[e] * EMBED + j], val,
                           __ATOMIC_RELAXED, __HIP_MEMORY_SCOPE_AGENT);
}

// ---------------------------------------------------------------------------
// 3) fused atom MLP: 64 rows per workgroup, 16 waves = 16 column tiles,
//    each wave holds 4 row-tiles of f32 WMMA accumulators (B-frag reuse x4)
// ---------------------------------------------------------------------------
__device__ __forceinline__ float swishf(float x) {
    return x * (1.0f / (1.0f + __expf(-x)));
}

__global__ __launch_bounds__(512) void atom_mlp_kernel(
    const float* __restrict__ summed,
    const float* __restrict__ W_up, const float* __restrict__ W_d1,
    const float* __restrict__ b_d1, const float* __restrict__ W_d2,
    const float* __restrict__ b_d2, const float* __restrict__ W_final,
    float*       __restrict__ out)
{
    extern __shared__ float smem[];
    float* ldsS  = smem;                                   // 64 x 129
    float* ldsH1 = smem  + ROWS_PER_WG * S_STRIDE;         // 64 x 257 (h1, later h3)
    float* ldsH2 = ldsH1 + ROWS_PER_WG * H_STRIDE;         // 64 x 257 (h2)

    const int tid   = threadIdx.x;
    const int lane  = tid & 31;
    const int wave  = tid >> 5;            // 0..15 -> column tile
    const int row0  = blockIdx.x * ROWS_PER_WG;

    // --- stage summed[64 x 128] into LDS (zero-pad past N) ---
    for (int p = tid; p < ROWS_PER_WG * EMBED; p += 512) {
        const int r = p >> 7, k = p & (EMBED - 1);
        const int gr = row0 + r;
        ldsS[r * S_STRIDE + k] = (gr < N_PARTICLES) ? summed[(long)gr * EMBED + k] : 0.f;
    }
    __syncthreads();

    const int m16  = lane & 15;            // row-in-tile (A) / col-in-tile (B,C,D)
    const int khi  = (lane >> 4) * 2;      // f32 WMMA: lanes 16..31 hold K+2,K+3
    const int vhi  = (lane >> 4) * 8;      // C/D: lanes 16..31 hold rows M+8..M+15
    const int col  = wave * 16 + m16;      // global output column 0..255

    v8f acc[4];

    // ================= layer 1: h1 = summed @ W_up  (K = 128) ===============
#pragma unroll
    for (int rt = 0; rt < 4; ++rt) acc[rt] = (v8f){};
    for (int k0 = 0; k0 < EMBED; k0 += 4) {
        const int kb = k0 + khi;
        v2f b;
        b.x = W_up[(kb    ) * OUT_EMBED + col];
        b.y = W_up[(kb + 1) * OUT_EMBED + col];
#pragma unroll
        for (int rt = 0; rt < 4; ++rt) {
            const float* ap = &ldsS[(rt * 16 + m16) * S_STRIDE + kb];
            v2f a; a.x = ap[0]; a.y = ap[1];
            acc[rt] = __builtin_amdgcn_wmma_f32_16x16x4_f32(
                false, a, false, b, (short)0, acc[rt], false, false);
        }
    }
#pragma unroll
    for (int rt = 0; rt < 4; ++rt)
#pragma unroll
        for (int v = 0; v < 8; ++v)
            ldsH1[(rt * 16 + vhi + v) * H_STRIDE + col] = acc[rt][v];
    __syncthreads();

    // ========== layer 2: h2 = swish(h1 @ W_d1 + b_d1)  (K = 256) ===========
#pragma unroll
    for (int rt = 0; rt < 4; ++rt) acc[rt] = (v8f){};
    for (int k0 = 0; k0 < OUT_EMBED; k0 += 4) {
        const int kb = k0 + khi;
        v2f b;
        b.x = W_d1[(kb    ) * OUT_EMBED + col];
        b.y = W_d1[(kb + 1) * OUT_EMBED + col];
#pragma unroll
        for (int rt = 0; rt < 4; ++rt) {
            const float* ap = &ldsH1[(rt * 16 + m16) * H_STRIDE + kb];
            v2f a; a.x = ap[0]; a.y = ap[1];
            acc[rt] = __builtin_amdgcn_wmma_f32_16x16x4_f32(
                false, a, false, b, (short)0, acc[rt], false, false);
        }
    }
    {
        const float bias = b_d1[col];
#pragma unroll
        for (int rt = 0; rt < 4; ++rt)
#pragma unroll
            for (int v = 0; v < 8; ++v)
                ldsH2[(rt * 16 + vhi + v) * H_STRIDE + col] = swishf(acc[rt][v] + bias);
    }
    __syncthreads();

    // ========== layer 3: h3 = swish(h2 @ W_d2 + b_d2)  (K = 256) ===========
#pragma unroll
    for (int rt = 0; rt < 4; ++rt) acc[rt] = (v8f){};
    for (int k0 = 0; k0 < OUT_EMBED; k0 += 4) {
        const int kb = k0 + khi;
        v2f b;
        b.x = W_d2[(kb    ) * OUT_EMBED + col];
        b.y = W_d2[(kb + 1) * OUT_EMBED + col];
#pragma unroll
        for (int rt = 0; rt < 4; ++rt) {
            const float* ap = &ldsH2[(rt * 16 + m16) * H_STRIDE + kb];
            v2f a; a.x = ap[0]; a.y = ap[1];
            acc[rt] = __builtin_amdgcn_wmma_f32_16x16x4_f32(
                false, a, false, b, (short)0, acc[rt], false, false);
        }
    }
    {
        const float bias = b_d2[col];
#pragma unroll
        for (int rt = 0; rt < 4; ++rt)
#pragma unroll
            for (int v = 0; v < 8; ++v)
                ldsH1[(rt * 16 + vhi + v) * H_STRIDE + col] = swishf(acc[rt][v] + bias); // h3 reuses H1
    }
    __syncthreads();

    // ============== final: out = h3 @ W_final  ([256] dot per row) ==========
    {
        const int r   = tid >> 3;     // 0..63
        const int seg = tid & 7;      // 8 threads per row
        float part = 0.f;
        const float* hp = &ldsH1[r * H_STRIDE + seg * 32];
        const float* wp = &W_final[seg * 32];
#pragma unroll 8
        for (int k = 0; k < 32; ++k) part = fmaf(hp[k], wp[k], part);
        part += __shfl_xor(part, 1, 32);
        part += __shfl_xor(part, 2, 32);
        part += __shfl_xor(part, 4, 32);
        const int gr = row0 + r;
        if (seg == 0 && gr < N_PARTICLES) out[gr] = part;
    }
}

// ---------------------------------------------------------------------------
extern "C" void kernel_launch(void* const* d_in, const int* in_sizes, int n_in,
                              void* d_out, int out_size, void* d_ws, size_t ws_size,
                              hipStream_t stream) {
    const float* messages = (const float*)d_in[0];
    const float* rbf      = (const float*)d_in[1];
    const int*   idx_i    = (const int*)  d_in[2];
    const float* W_rbf    = (const float*)d_in[3];
    const float* W_up     = (const float*)d_in[4];
    const float* W_d1     = (const float*)d_in[5];
    const float* b_d1     = (const float*)d_in[6];
    const float* W_d2     = (const float*)d_in[7];
    const float* b_d2     = (const float*)d_in[8];
    const float* W_final  = (const float*)d_in[9];
    float* out    = (float*)d_out;
    float* summed = (float*)d_ws;                     // 50000 x 128 f32 = 25.6 MB

    const int n4 = (N_PARTICLES * EMBED) / 4;
    zero_kernel<<<(n4 + 255) / 256, 256, 0, stream>>>((float4*)summed, n4);

    edge_scatter_kernel<<<N_EDGES / 2, 256, 0, stream>>>(
        messages, rbf, idx_i, W_rbf, summed);

    const int nblk = (N_PARTICLES + ROWS_PER_WG - 1) / ROWS_PER_WG;   // 782
    const size_t lds_bytes =
        (size_t)(ROWS_PER_WG * S_STRIDE + 2 * ROWS_PER_WG * H_STRIDE) * sizeof(float); // 164,608 B
    atom_mlp_kernel<<<nblk, 512, lds_bytes, stream>>>(
        summed, W_up, W_d1, b_d1, W_d2, b_d2, W_final, out);
}